// NMSWithOnnxSupport_26706106647080
// MI455X (gfx1250) — compile-verified
//
#include <hip/hip_runtime.h>
#include <hip/hip_bf16.h>

#define N_BOXES   20000
#define N_CLASSES 80
#define TOPK      1000
#define KPAD      1008          // 63 tiles of 16
#define MAXPC     200
#define MAXPI     100
#define CONF_T    0.05f
#define NMS_T     0.5f
#define NBINS     8192
#define CAND_CAP  2048
#define GSORT     16384

typedef float v2f __attribute__((ext_vector_type(2)));
typedef float v8f __attribute__((ext_vector_type(8)));

// ---------------------------------------------------------------------------
// Kernel 1: per-class top-1000 (sorted desc, index ties ascending like top_k).
// Histogram radix-select on float bits, compact candidates, bitonic sort.
// ---------------------------------------------------------------------------
__global__ __launch_bounds__(256) void topk_per_class(
    const float* __restrict__ scores,   // (N_BOXES, N_CLASSES) row-major
    float* __restrict__ topS,           // (N_CLASSES, TOPK)
    int* __restrict__ topIdx)           // (N_CLASSES, TOPK)
{
    __shared__ unsigned hist[NBINS];                    // 32 KB
    __shared__ unsigned long long cand[CAND_CAP];       // 16 KB
    __shared__ unsigned chunkSum[256];
    __shared__ unsigned s_bThr;
    __shared__ unsigned s_cnt;

    const int c   = blockIdx.x;
    const int tid = threadIdx.x;

    for (int b = tid; b < NBINS; b += 256) hist[b] = 0u;
    if (tid == 0) s_cnt = 0u;
    __syncthreads();

    // Pass 1: histogram of top 13 bits of the (non-negative) float pattern.
    for (int n = tid; n < N_BOXES; n += 256) {
        const float* p = scores + (size_t)n * N_CLASSES + c;
        if (n + 2048 < N_BOXES)
            __builtin_prefetch(scores + (size_t)(n + 2048) * N_CLASSES + c, 0, 1);
        unsigned bits = __float_as_uint(*p);
        atomicAdd(&hist[bits >> 19], 1u);
    }
    __syncthreads();

    // Find threshold bin (cumulative from the top crosses TOPK).
    unsigned csum = 0;
    for (int b = 0; b < 32; ++b) csum += hist[tid * 32 + b];
    chunkSum[tid] = csum;
    __syncthreads();
    if (tid == 0) {
        unsigned cum = 0; int ch = 255;
        for (; ch >= 0; --ch) {
            if (cum + chunkSum[ch] >= TOPK) break;
            cum += chunkSum[ch];
        }
        if (ch < 0) ch = 0;
        unsigned bThr = (unsigned)(ch * 32);
        for (int b = 31; b >= 0; --b) {
            unsigned h = hist[ch * 32 + b];
            if (cum + h >= TOPK) { bThr = (unsigned)(ch * 32 + b); break; }
            cum += h;
        }
        s_bThr = bThr;
    }
    __syncthreads();
    const unsigned bThr = s_bThr;

    // Pass 2: compact candidates >= threshold bin as 64-bit sort keys.
    // key = scoreBits<<32 | (~idx) -> desc sort gives desc score, asc index.
    for (int n = tid; n < N_BOXES; n += 256) {
        unsigned bits = __float_as_uint(scores[(size_t)n * N_CLASSES + c]);
        if ((bits >> 19) >= bThr) {
            unsigned pos = atomicAdd(&s_cnt, 1u);
            if (pos < CAND_CAP)
                cand[pos] = ((unsigned long long)bits << 32) |
                            (unsigned long long)(0xFFFFFFFFu - (unsigned)n);
        }
    }
    __syncthreads();
    unsigned cnt = s_cnt; if (cnt > CAND_CAP) cnt = CAND_CAP;
    for (int p = (int)cnt + tid; p < CAND_CAP; p += 256) cand[p] = 0ull;
    __syncthreads();

    // Bitonic sort, descending, 2048 keys.
    for (int k = 2; k <= CAND_CAP; k <<= 1) {
        for (int j = k >> 1; j > 0; j >>= 1) {
            for (int p = tid; p < CAND_CAP / 2; p += 256) {
                int i   = ((p & ~(j - 1)) << 1) | (p & (j - 1));
                int ixj = i | j;
                unsigned long long a = cand[i], b = cand[ixj];
                if ((((i & k) == 0) ? (a < b) : (a > b))) {
                    cand[i] = b; cand[ixj] = a;
                }
            }
            __syncthreads();
        }
    }

    for (int r = tid; r < TOPK; r += 256) {
        unsigned long long key = cand[r];
        topS[c * TOPK + r]   = __uint_as_float((unsigned)(key >> 32));
        topIdx[c * TOPK + r] = (int)(0xFFFFFFFFu - (unsigned)(key & 0xFFFFFFFFu));
    }
}

// ---------------------------------------------------------------------------
// Kernel 2: per-class NMS. WMMA-tiled adjacency bitmask in LDS, then a
// single-wave sequential greedy suppression over 1024-bit rows, then compact
// surviving (already sorted) scores into the per-class top-200 slots.
// ---------------------------------------------------------------------------
__global__ __launch_bounds__(256) void nms_per_class(
    const float* __restrict__ boxes,    // (N_BOXES, 4)
    const float* __restrict__ topS,
    const int* __restrict__ topIdx,
    float* __restrict__ kept)           // (N_CLASSES, MAXPC)
{
    extern __shared__ char smem[];
    float*    sx1 = (float*)smem;            // each array 16B-aligned (KPAD*4 = 4032)
    float*    sy1 = sx1 + KPAD;
    float*    sx2 = sy1 + KPAD;
    float*    sy2 = sx2 + KPAD;
    float*    sar = sy2 + KPAD;
    float*    ssc = sar + KPAD;
    unsigned* adj = (unsigned*)(ssc + KPAD);     // KPAD rows x 32 dwords
    unsigned* kb  = adj + KPAD * 32;             // 32-dword keep bitmask
    unsigned* wp  = kb + 32;                     // 33-entry popcount prefix

    const int c   = blockIdx.x;
    const int tid = threadIdx.x;

    for (int r = tid; r < MAXPC; r += 256) kept[c * MAXPC + r] = 0.0f;

    // Gather top-1000 boxes into LDS (padded rows are degenerate: iou == 0).
    for (int j = tid; j < KPAD; j += 256) {
        float x1 = 0.f, y1 = 0.f, x2 = 0.f, y2 = 0.f, s = 0.f;
        if (j < TOPK) {
            s = topS[c * TOPK + j];
            int bi = topIdx[c * TOPK + j];
            const float* bp = boxes + (size_t)bi * 4;
            x1 = bp[0]; y1 = bp[1]; x2 = bp[2]; y2 = bp[3];
        }
        sx1[j] = x1; sy1[j] = y1; sx2[j] = x2; sy2[j] = y2;
        sar[j] = (x2 - x1) * (y2 - y1);
        ssc[j] = s;
    }
    for (int w = tid; w < KPAD * 32; w += 256) adj[w] = 0u;
    __syncthreads();

    // keep init: score > CONF_T
    if (tid < 32) {
        unsigned word = 0u;
        for (int b = 0; b < 32; ++b) {
            int j = tid * 32 + b;
            if (j < TOPK && ssc[j] > CONF_T) word |= (1u << b);
        }
        kb[tid] = word;
    }

    // Phase 1: adjacency matrix, 16x16 tiles per wave.
    // area_i + area_j via V_WMMA_F32_16X16X4_F32 (rank-2 outer-sum on the
    // matrix pipe), intersection via VALU, ballot packs bits. The IoU>0.5
    // predicate is evaluated division-free: inter/(sum-inter+eps) > 0.5
    // <=> 3*inter > sum + eps (denominator strictly positive).
    const int lane = threadIdx.x & 31;
    const int wave = threadIdx.x >> 5;
    const int NT   = KPAD / 16;   // 63
    for (int t = wave; t < NT * NT; t += 8) {
        int ti = t / NT, tj = t % NT;
        int iBase = ti * 16, jBase = tj * 16;
        int n = jBase + (lane & 15);
        float x1j = sx1[n], y1j = sy1[n], x2j = sx2[n], y2j = sy2[n];
        float aj  = sar[n];
        bool lo = lane < 16;
        int  mi = iBase + (lane & 15);
        v2f a, b;
        a.x = lo ? sar[mi] : 0.0f;   // A[M][K=0] = area_i   (lanes 16-31: K=2,3 -> 0)
        a.y = lo ? 1.0f    : 0.0f;   // A[M][K=1] = 1
        b.x = lo ? 1.0f    : 0.0f;   // B[K=0][N] = 1
        b.y = lo ? aj      : 0.0f;   // B[K=1][N] = area_j
        v8f cacc = {};
        v8f sum = __builtin_amdgcn_wmma_f32_16x16x4_f32(
            /*neg_a=*/false, a, /*neg_b=*/false, b,
            /*c_mod=*/(short)0, cacc, /*reuse_a=*/false, /*reuse_b=*/false);

        int mOff = iBase + ((lane >> 4) << 3);  // D: lanes 0-15 -> M=r, 16-31 -> M=r+8
        // Hoist the 8 row values per column array as two b128 LDS loads each.
        float rx1[8], ry1[8], rx2[8], ry2[8];
        *(float4*)&rx1[0] = *(const float4*)&sx1[mOff];
        *(float4*)&rx1[4] = *(const float4*)&sx1[mOff + 4];
        *(float4*)&ry1[0] = *(const float4*)&sy1[mOff];
        *(float4*)&ry1[4] = *(const float4*)&sy1[mOff + 4];
        *(float4*)&rx2[0] = *(const float4*)&sx2[mOff];
        *(float4*)&rx2[4] = *(const float4*)&sx2[mOff + 4];
        *(float4*)&ry2[0] = *(const float4*)&sy2[mOff];
        *(float4*)&ry2[4] = *(const float4*)&sy2[mOff + 4];

        int w  = tj >> 1;
        int sh = (tj & 1) * 16;
#pragma unroll
        for (int r = 0; r < 8; ++r) {
            float iw = fminf(rx2[r], x2j) - fmaxf(rx1[r], x1j);
            float ih = fminf(ry2[r], y2j) - fmaxf(ry1[r], y1j);
            iw = fmaxf(iw, 0.0f); ih = fmaxf(ih, 0.0f);
            float inter = iw * ih;
            bool sup = (3.0f * inter) > (sum[r] + 1e-9f);   // iou > 0.5
            unsigned bal = __builtin_amdgcn_ballot_w32(sup);
            if (lane == 0) {
                atomicOr(&adj[(iBase + r) * 32 + w],     (bal & 0xFFFFu) << sh);
                atomicOr(&adj[(iBase + 8 + r) * 32 + w], (bal >> 16)     << sh);
            }
        }
    }
    __syncthreads();

    // Phase 2: sequential greedy suppression, wave 0 only.
    // Lane L owns keep word L (bits 32L..32L+31). Same-wave LDS is in-order.
    if (tid < 32) {
        for (int i = 0; i < TOPK; ++i) {
            unsigned kw = kb[i >> 5];
            if ((kw >> (i & 31)) & 1u) {
                unsigned adjw = adj[i * 32 + tid];
                int iw2 = i >> 5, ib = i & 31;
                unsigned mask;
                if (tid > iw2)      mask = 0xFFFFFFFFu;          // all j > i
                else if (tid < iw2) mask = 0u;                   // all j < i
                else mask = (ib == 31) ? 0u : (0xFFFFFFFFu << (ib + 1));
                kb[tid] &= ~(adjw & mask);
            }
        }
    }
    __syncthreads();

    // Phase 3: compact survivors in (sorted) order -> first MAXPC scores.
    if (tid == 0) {
        unsigned acc = 0;
        for (int w2 = 0; w2 < 32; ++w2) { wp[w2] = acc; acc += __popc(kb[w2]); }
        wp[32] = acc;
    }
    __syncthreads();
    for (int j = tid; j < TOPK; j += 256) {
        unsigned word = kb[j >> 5];
        if ((word >> (j & 31)) & 1u) {
            unsigned rank = wp[j >> 5] + __popc(word & ((1u << (j & 31)) - 1u));
            if (rank < MAXPC) kept[c * MAXPC + rank] = ssc[j];
        }
    }
}

// ---------------------------------------------------------------------------
// Kernel 3: global top-100 of the 16000 kept scores (sorted desc).
// ---------------------------------------------------------------------------
__global__ __launch_bounds__(1024) void global_topk(
    const float* __restrict__ kept, float* __restrict__ out)
{
    extern __shared__ char smem[];
    float* s = (float*)smem;
    const int tid = threadIdx.x;
    for (int i = tid; i < GSORT; i += 1024)
        s[i] = (i < N_CLASSES * MAXPC) ? kept[i] : -1.0f;
    __syncthreads();
    for (int k = 2; k <= GSORT; k <<= 1) {
        for (int j = k >> 1; j > 0; j >>= 1) {
            for (int p = tid; p < GSORT / 2; p += 1024) {
                int i   = ((p & ~(j - 1)) << 1) | (p & (j - 1));
                int ixj = i | j;
                float a = s[i], b = s[ixj];
                if ((((i & k) == 0) ? (a < b) : (a > b))) {
                    s[i] = b; s[ixj] = a;
                }
            }
            __syncthreads();
        }
    }
    for (int r = tid; r < MAXPI; r += 1024) out[r] = s[r];
}

// ---------------------------------------------------------------------------
extern "C" void kernel_launch(void* const* d_in, const int* in_sizes, int n_in,
                              void* d_out, int out_size, void* d_ws, size_t ws_size,
                              hipStream_t stream) {
    (void)in_sizes; (void)n_in; (void)out_size; (void)ws_size;
    const float* scores = (const float*)d_in[0];   // (1, 20000, 80) fp32
    const float* boxes  = (const float*)d_in[1];   // (20000, 4)     fp32
    float* out = (float*)d_out;                    // (100,)         fp32

    char* ws = (char*)d_ws;
    float* topS   = (float*)ws;                                        // 320000 B
    int*   topIdx = (int*)(ws + (size_t)N_CLASSES * TOPK * 4);         // 320000 B
    float* kept   = (float*)(ws + 2 * (size_t)N_CLASSES * TOPK * 4);   //  64000 B

    const size_t smemB = (size_t)(6 * KPAD) * 4      // box columns + scores
                       + (size_t)KPAD * 32 * 4       // adjacency bitmask (126 KB)
                       + 32 * 4 + 33 * 4;            // keep mask + prefix
    // Opt in to >64KB dynamic LDS (WGP has 320KB on gfx1250).
    (void)hipFuncSetAttribute((const void*)nms_per_class,
                              hipFuncAttributeMaxDynamicSharedMemorySize,
                              (int)smemB);

    topk_per_class<<<N_CLASSES, 256, 0, stream>>>(scores, topS, topIdx);
    nms_per_class<<<N_CLASSES, 256, smemB, stream>>>(boxes, topS, topIdx, kept);
    global_topk<<<1, 1024, (size_t)GSORT * 4, stream>>>(kept, out);
}